// MessagePassingBlock_78460462563621
// MI455X (gfx1250) — compile-verified
//
#include <hip/hip_runtime.h>
#include <hip/hip_bf16.h>

typedef __attribute__((ext_vector_type(2))) float v2f;
typedef __attribute__((ext_vector_type(8))) float v8f;

#define DIM 128

// Fused dual GEMM: h = x @ W_msg ; acc = x @ W_self + b
// V_WMMA_F32_16X16X4_F32 (exact fp32). Each wave computes a 32x16 output patch:
// two 16-row M-tiles sharing the same B fragments (4 WMMAs per B load pair).
// A-fragment (16x4 f32, 2 VGPRs): lane l(0-15): M=l, K=kk+0/kk+1 ; lane l+16: M=l, K=kk+2/kk+3.
// B-fragment (4x16 f32, 2 VGPRs): lane l(0-15): N=l, K=kk+0/kk+1 ; lane l+16: N=l, K=kk+2/kk+3.
// C/D (8 VGPRs): row = m0 + r + 8*(lane>>4), col = n0 + (lane&15).
// Requires N % 32 == 0 (N = 100000 = 3125 * 32): no guards, EXEC all-1s.
__global__ void __launch_bounds__(256) rgcn_dual_gemm(
    const float* __restrict__ x,  const float* __restrict__ Wm,
    const float* __restrict__ Ws, const float* __restrict__ bias,
    float* __restrict__ h, float* __restrict__ acc)
{
    const int lane = threadIdx.x & 31;
    const int wv   = threadIdx.x >> 5;      // 0..7 -> N-tile
    const int m0   = blockIdx.x * 32;       // two 16-row tiles: m0, m0+16
    const int n0   = wv * 16;
    const int mr   = lane & 15;
    const int hi   = lane >> 4;             // 0 or 1
    const int kh   = hi * 2;
    const int ncol = n0 + mr;

    const float* xrow0 = x + (long)(m0 + mr) * DIM + kh;        // M-tile 0
    const float* xrow1 = xrow0 + 16 * DIM;                      // M-tile 1

    v8f cm0 = {}, cs0 = {};   // tile 0: x@W_msg, x@W_self
    v8f cm1 = {}, cs1 = {};   // tile 1
    #pragma unroll 4
    for (int kk = 0; kk < DIM; kk += 4) {
        const int ka = kk + kh;
        v2f a0 = *(const v2f*)(xrow0 + kk);
        v2f a1 = *(const v2f*)(xrow1 + kk);
        v2f bm, bs;
        bm.x = Wm[(long)ka * DIM + ncol];
        bm.y = Wm[(long)(ka + 1) * DIM + ncol];
        bs.x = Ws[(long)ka * DIM + ncol];
        bs.y = Ws[(long)(ka + 1) * DIM + ncol];
        cm0 = __builtin_amdgcn_wmma_f32_16x16x4_f32(false, a0, false, bm, (short)0, cm0, false, false);
        cm1 = __builtin_amdgcn_wmma_f32_16x16x4_f32(false, a1, false, bm, (short)0, cm1, false, false);
        cs0 = __builtin_amdgcn_wmma_f32_16x16x4_f32(false, a0, false, bs, (short)0, cs0, false, false);
        cs1 = __builtin_amdgcn_wmma_f32_16x16x4_f32(false, a1, false, bs, (short)0, cs1, false, false);
    }

    const float bn = bias[ncol];
    const long  cbase = (long)(m0 + 8 * hi) * DIM + ncol;
    #pragma unroll
    for (int r = 0; r < 8; ++r) {
        const long o0 = cbase + (long)r * DIM;          // tile 0 row
        const long o1 = o0 + 16 * DIM;                  // tile 1 row
        h[o0]   = cm0[r];
        h[o1]   = cm1[r];
        acc[o0] = cs0[r] + bn;
        acc[o1] = cs1[r] + bn;
    }
}

// One wave per edge; each lane owns 4 consecutive channels (float4 = 16B/lane,
// 512B fully-coalesced per wave). Gather from h (51MB, L2-resident in 192MB L2),
// scatter with hardware fp32 atomics (no-return -> STOREcnt path).
// E = 625000 = 78125 * 8: exact grid, no tail.
__global__ void __launch_bounds__(256) rgcn_edge_scatter(
    const float* __restrict__ h,  const int* __restrict__ src,
    const int* __restrict__ tgt,  const int* __restrict__ et,
    const float* __restrict__ ew, const float* __restrict__ rb,
    float* __restrict__ acc, int E)
{
    const int lane = threadIdx.x & 31;
    const long e = (long)((blockIdx.x * blockDim.x + threadIdx.x) >> 5);
    if (e >= E) return;

    const int   s = src[e];
    const int   t = tgt[e];
    const int   r = et[e];
    const float w = ew[e];

    const float4 hv = *(const float4*)(h  + (long)s * DIM + lane * 4);
    const float4 rv = *(const float4*)(rb + (long)r * DIM + lane * 4);

    float* dst = acc + (long)t * DIM + lane * 4;
    __hip_atomic_fetch_add(dst + 0, (hv.x + rv.x) * w, __ATOMIC_RELAXED, __HIP_MEMORY_SCOPE_AGENT);
    __hip_atomic_fetch_add(dst + 1, (hv.y + rv.y) * w, __ATOMIC_RELAXED, __HIP_MEMORY_SCOPE_AGENT);
    __hip_atomic_fetch_add(dst + 2, (hv.z + rv.z) * w, __ATOMIC_RELAXED, __HIP_MEMORY_SCOPE_AGENT);
    __hip_atomic_fetch_add(dst + 3, (hv.w + rv.w) * w, __ATOMIC_RELAXED, __HIP_MEMORY_SCOPE_AGENT);
}

// In-place ReLU epilogue over the accumulated output.
__global__ void __launch_bounds__(256) rgcn_relu(float* __restrict__ out, long total4)
{
    const long i = (long)blockIdx.x * blockDim.x + threadIdx.x;
    if (i >= total4) return;
    float4 v = *(float4*)(out + i * 4);
    v.x = fmaxf(v.x, 0.0f);
    v.y = fmaxf(v.y, 0.0f);
    v.z = fmaxf(v.z, 0.0f);
    v.w = fmaxf(v.w, 0.0f);
    *(float4*)(out + i * 4) = v;
}

extern "C" void kernel_launch(void* const* d_in, const int* in_sizes, int n_in,
                              void* d_out, int out_size, void* d_ws, size_t ws_size,
                              hipStream_t stream) {
    const float* x    = (const float*)d_in[0];   // (N, 128)
    const int*   src  = (const int*)  d_in[1];   // (E,)
    const int*   tgt  = (const int*)  d_in[2];   // (E,)
    const int*   et   = (const int*)  d_in[3];   // (E,)
    const float* ew   = (const float*)d_in[4];   // (E,)
    const float* Wm   = (const float*)d_in[5];   // (128, 128)
    const float* rb   = (const float*)d_in[6];   // (8, 128)
    const float* Ws   = (const float*)d_in[7];   // (128, 128)
    const float* bias = (const float*)d_in[8];   // (128,)

    float* out = (float*)d_out;                  // (N, 128) pre-ReLU accumulator -> output
    float* h   = (float*)d_ws;                   // (N, 128) scratch: x @ W_msg

    const int N = in_sizes[0] / DIM;             // 100000
    const int E = in_sizes[1];                   // 625000

    // 1) fused GEMMs: h = x@W_msg ; out = x@W_self + b   (N % 32 == 0)
    rgcn_dual_gemm<<<N / 32, 256, 0, stream>>>(x, Wm, Ws, bias, h, out);

    // 2) edge gather + scale + atomic scatter-sum into out
    rgcn_edge_scatter<<<(E + 7) / 8, 256, 0, stream>>>(h, src, tgt, et, ew, rb, out, E);

    // 3) in-place ReLU
    const long total4 = (long)N * DIM / 4;
    rgcn_relu<<<(int)((total4 + 255) / 256), 256, 0, stream>>>(out, total4);
}